// GATBackbone_15135464751820
// MI455X (gfx1250) — compile-verified
//
#include <hip/hip_runtime.h>
#include <math.h>

// GATv2 backbone on MI455X (gfx1250, wave32).
// fp32 WMMA (v_wmma_f32_16x16x4_f32) for all GEMMs; fused edge pipeline.

#define NN   50000
#define EE   800000
#define DD   64
#define HH   4
#define CC   16
#define EDIM 16
#define LL   3
#define NEG_SLOPE 0.2f

typedef float v2f __attribute__((ext_vector_type(2)));
typedef float v8f __attribute__((ext_vector_type(8)));

// ---- float atomic max via signed/unsigned int ordering trick ----
__device__ inline void atomicMaxFloat(float* addr, float val) {
    if (val >= 0.0f) {
        atomicMax((int*)addr, __float_as_int(val));
    } else {
        atomicMin((unsigned int*)addr, __float_as_uint(val));
    }
}

// ---------------------------------------------------------------------------
// init: accum[N*D] = 0, smax[N*H] = -inf, denom[N*H] = 0
// ---------------------------------------------------------------------------
__global__ __launch_bounds__(256) void gat_init(float* __restrict__ accum,
                                                float* __restrict__ smax,
                                                float* __restrict__ denom) {
    int i = blockIdx.x * 256 + threadIdx.x;          // grid covers N*D
    accum[i] = 0.0f;
    if (i < NN * HH) {
        smax[i]  = __int_as_float(0xff800000);       // -inf
        denom[i] = 0.0f;
    }
}

// ---------------------------------------------------------------------------
// node_xform: xl = h@Wl + bl ; xr = h@Wr + br  (both [N,64])
// Block = 256 threads = 8 waves. Waves 0-3 -> xl col-tiles 0..3,
// waves 4-7 -> xr col-tiles 0..3. Row tile = 16 nodes per block.
// f32 WMMA 16x16x4, K-loop of 16 steps.
// A layout (ISA 7.12.2): lane<16 holds A[M=lane][k0,k0+1], lane>=16 holds K=k0+2,k0+3.
// B layout: VGPR0 = row k0 (lanes 0-15) / row k0+2 (lanes 16-31); VGPR1 = +1.
// D layout: D[g + 8*(lane>>4)][lane&15] in acc[g].
// ---------------------------------------------------------------------------
__global__ __launch_bounds__(256) void node_xform(const float* __restrict__ h,
                                                  const float* __restrict__ Wl,
                                                  const float* __restrict__ bl,
                                                  const float* __restrict__ Wr,
                                                  const float* __restrict__ br,
                                                  float* __restrict__ xl,
                                                  float* __restrict__ xr) {
    const int wave    = threadIdx.x >> 5;
    const int lane    = threadIdx.x & 31;
    const int row0    = blockIdx.x * 16;
    const int col0    = (wave & 3) * 16;
    const bool right  = wave >= 4;
    const float* W    = right ? Wr : Wl;
    const float* bvec = right ? br : bl;
    float* out        = right ? xr : xl;

    const int halfsel = lane >> 4;       // 0 | 1
    const int l16     = lane & 15;

    v8f acc = {};
    const float* arow = h + (size_t)(row0 + l16) * DD;
#pragma unroll
    for (int k0 = 0; k0 < DD; k0 += 4) {
        v2f a = *(const v2f*)(arow + k0 + 2 * halfsel);
        v2f b;
        b.x = W[(k0 + 2 * halfsel)     * DD + col0 + l16];
        b.y = W[(k0 + 2 * halfsel + 1) * DD + col0 + l16];
        acc = __builtin_amdgcn_wmma_f32_16x16x4_f32(false, a, false, b,
                                                    (short)0, acc, false, false);
    }
    const float bias = bvec[col0 + l16];
#pragma unroll
    for (int g = 0; g < 8; ++g) {
        int m = g + 8 * halfsel;
        out[(size_t)(row0 + m) * DD + col0 + l16] = acc[g] + bias;
    }
}

// ---------------------------------------------------------------------------
// edge_score: per wave -> (16-edge tile, head).
//   ef = EA[16x16] @ We[:, head*16 .. head*16+16)   (4 f32 WMMAs)
//   m  = leaky_relu(ef + xl[src] + xr[dst])
//   score[e,h] = sum_c m*att[h,c]  (16-lane xor-shuffle reduce)
//   atomic max into smax[dst,h]
// Block = 8 waves = 2 edge tiles x 4 heads. Grid = E/32.
// ---------------------------------------------------------------------------
__global__ __launch_bounds__(256) void edge_score(const float* __restrict__ ea,
                                                  const int*   __restrict__ src,
                                                  const int*   __restrict__ dst,
                                                  const float* __restrict__ We,
                                                  const float* __restrict__ att,
                                                  const float* __restrict__ xl,
                                                  const float* __restrict__ xr,
                                                  float* __restrict__ score,
                                                  float* __restrict__ smax) {
    const int wave = threadIdx.x >> 5;
    const int lane = threadIdx.x & 31;
    const int tile = blockIdx.x * 2 + (wave >> 2);
    const int hth  = wave & 3;                       // head index
    const int e0   = tile * 16;
    const int halfsel = lane >> 4;
    const int l16     = lane & 15;

    // prefetch next tile's edge_attr (streams once, 51MB/layer)
    if (e0 + 16 < EE)
        __builtin_prefetch(ea + (size_t)(e0 + 16 + l16) * EDIM, 0, 1);

    v8f acc = {};
    const float* earow = ea + (size_t)(e0 + l16) * EDIM;
#pragma unroll
    for (int k0 = 0; k0 < EDIM; k0 += 4) {
        v2f a = *(const v2f*)(earow + k0 + 2 * halfsel);
        v2f b;
        b.x = We[(k0 + 2 * halfsel)     * DD + hth * CC + l16];
        b.y = We[(k0 + 2 * halfsel + 1) * DD + hth * CC + l16];
        acc = __builtin_amdgcn_wmma_f32_16x16x4_f32(false, a, false, b,
                                                    (short)0, acc, false, false);
    }

    const float av   = att[hth * CC + l16];
    const int   colg = hth * CC + l16;               // column within D
#pragma unroll
    for (int g = 0; g < 8; ++g) {
        const int e = e0 + g + 8 * halfsel;
        const int s = src[e];
        const int d = dst[e];
        // node features are L2-resident (12.8MB each) -> cheap gathers
        float m = acc[g] + xl[(size_t)s * DD + colg] + xr[(size_t)d * DD + colg];
        m = (m > 0.0f) ? m : NEG_SLOPE * m;
        float v = m * av;
        // reduce over the 16 channels (lanes within each half-group)
#pragma unroll
        for (int off = 8; off >= 1; off >>= 1)
            v += __shfl_xor(v, off, 32);
        if (l16 == 0) {
            score[(size_t)e * HH + hth] = v;
            atomicMaxFloat(&smax[(size_t)d * HH + hth], v);
        }
    }
}

// ---------------------------------------------------------------------------
// edge_expdenom: ex = exp(score - smax[dst]); denom[dst] += ex  (thread per E*H)
// ---------------------------------------------------------------------------
__global__ __launch_bounds__(256) void edge_expdenom(const int* __restrict__ dst,
                                                     const float* __restrict__ smax,
                                                     float* __restrict__ score,
                                                     float* __restrict__ denom) {
    int i = blockIdx.x * 256 + threadIdx.x;          // grid covers E*H exactly
    int e = i >> 2;
    int h = i & 3;
    int d = dst[e];
    float ex = __expf(score[i] - smax[(size_t)d * HH + h]);
    score[i] = ex;                                   // reuse as ex buffer
    atomicAdd(&denom[(size_t)d * HH + h], ex);
}

// ---------------------------------------------------------------------------
// edge_scatter: accum[dst,col] += (ex/denom[dst,h]) * xl[src,col]
// thread per (edge, channel); E*64 threads.
// ---------------------------------------------------------------------------
__global__ __launch_bounds__(256) void edge_scatter(const int* __restrict__ src,
                                                    const int* __restrict__ dst,
                                                    const float* __restrict__ ex,
                                                    const float* __restrict__ denom,
                                                    const float* __restrict__ xl,
                                                    float* __restrict__ accum) {
    long long i = (long long)blockIdx.x * 256 + threadIdx.x;  // covers E*64 exactly
    int e   = (int)(i >> 6);
    int col = (int)(i & 63);
    int h   = col >> 4;
    int s   = src[e];
    int d   = dst[e];
    float alpha = ex[(size_t)e * HH + h] / denom[(size_t)d * HH + h];
    atomicAdd(&accum[(size_t)d * DD + col], alpha * xl[(size_t)s * DD + col]);
}

// ---------------------------------------------------------------------------
// finalize: hout = relu(accum + bias)   (thread per N*D)
// ---------------------------------------------------------------------------
__global__ __launch_bounds__(256) void gat_finalize(const float* __restrict__ accum,
                                                    const float* __restrict__ bias,
                                                    float* __restrict__ hout) {
    int i = blockIdx.x * 256 + threadIdx.x;          // grid covers N*D exactly
    float v = accum[i] + bias[i & 63];
    hout[i] = v > 0.0f ? v : 0.0f;
}

// ---------------------------------------------------------------------------
extern "C" void kernel_launch(void* const* d_in, const int* in_sizes, int n_in,
                              void* d_out, int out_size, void* d_ws, size_t ws_size,
                              hipStream_t stream) {
    const float* x    = (const float*)d_in[0];
    const int*   eidx = (const int*)  d_in[1];
    const float* ea   = (const float*)d_in[2];
    const float* Wl   = (const float*)d_in[3];
    const float* bl   = (const float*)d_in[4];
    const float* Wr   = (const float*)d_in[5];
    const float* br   = (const float*)d_in[6];
    const float* We   = (const float*)d_in[7];
    const float* att  = (const float*)d_in[8];
    const float* bias = (const float*)d_in[9];
    float* hout = (float*)d_out;

    const int* src = eidx;
    const int* dst = eidx + EE;

    // workspace carve (~52 MB)
    float* xl    = (float*)d_ws;                 // N*D
    float* xr    = xl    + (size_t)NN * DD;      // N*D
    float* score = xr    + (size_t)NN * DD;      // E*H  (score, then ex)
    float* smax  = score + (size_t)EE * HH;      // N*H
    float* denom = smax  + (size_t)NN * HH;      // N*H
    float* accum = denom + (size_t)NN * HH;      // N*D

    const int gNodeElem = (NN * DD) / 256;       // 12500
    const int gRowTiles = NN / 16;               // 3125
    const int gEdgeTile = EE / 32;               // 25000
    const int gEdgeHead = (EE * HH) / 256;       // 12500
    const int gEdgeChan = (int)(((long long)EE * DD) / 256);  // 200000

    for (int l = 0; l < LL; ++l) {
        const float* hin = (l == 0) ? x : hout;
        gat_init<<<gNodeElem, 256, 0, stream>>>(accum, smax, denom);
        node_xform<<<gRowTiles, 256, 0, stream>>>(hin,
                                                  Wl + (size_t)l * DD * DD,
                                                  bl + (size_t)l * DD,
                                                  Wr + (size_t)l * DD * DD,
                                                  br + (size_t)l * DD,
                                                  xl, xr);
        edge_score<<<gEdgeTile, 256, 0, stream>>>(ea, src, dst,
                                                  We  + (size_t)l * EDIM * DD,
                                                  att + (size_t)l * HH * CC,
                                                  xl, xr, score, smax);
        edge_expdenom<<<gEdgeHead, 256, 0, stream>>>(dst, smax, score, denom);
        edge_scatter<<<gEdgeChan, 256, 0, stream>>>(src, dst, score, denom, xl, accum);
        gat_finalize<<<gNodeElem, 256, 0, stream>>>(accum, bias + (size_t)l * DD, hout);
    }
}